// HippocampalInhibitoryNetwork_27960237097631
// MI455X (gfx1250) — compile-verified
//
#include <hip/hip_runtime.h>
#include <hip/hip_bf16.h>

// ---------------------------------------------------------------------------
// HippocampalInhibitoryNetwork on MI455X (gfx1250, wave32)
//
// Cost model: W_gap (14000x14000 fp32 = 784 MB) streamed once ->
//   ~34 us at 23.3 TB/s; everything else is noise. We fuse BOTH per-row
//   reductions (row-sum and W@V) into one streaming pass using
//   V_WMMA_F32_16X16X4_F32: A = 16x4 tile of W, B column0 = V[k..k+3],
//   B column1 = 1.0 -> C[:,0] accumulates dot, C[:,1] accumulates rowsum.
// ---------------------------------------------------------------------------

#define N_PV   14000
#define N_OLM  12000
#define N_BIST 8000
#define N_OUT  (N_PV + N_OLM + N_BIST)

#define E_E_C    3.0f
#define E_I_C   (-0.5f)
#define E_NMDA_C 3.0f
#define E_L_C    0.0f

typedef __attribute__((ext_vector_type(2))) float v2f;
typedef __attribute__((ext_vector_type(8))) float v8f;

// ---------------------------------------------------------------------------
// Kernel 1: drive = mean(pv_g_exc), single block reduction.
// ---------------------------------------------------------------------------
__global__ __launch_bounds__(256) void reduce_drive_kernel(
    const float* __restrict__ g, float* __restrict__ out) {
  __shared__ float sdata[256];
  float s = 0.0f;
  for (int i = threadIdx.x; i < N_PV; i += 256) s += g[i];
  sdata[threadIdx.x] = s;
  __syncthreads();
  for (int ofs = 128; ofs > 0; ofs >>= 1) {
    if ((int)threadIdx.x < ofs) sdata[threadIdx.x] += sdata[threadIdx.x + ofs];
    __syncthreads();
  }
  if (threadIdx.x == 0) out[0] = sdata[0] * (1.0f / (float)N_PV);
}

// ---------------------------------------------------------------------------
// Kernel 2: fused GEMV + row-sum over W_gap via V_WMMA_F32_16X16X4_F32.
//   grid.x = 875 (16 rows each), block = 256 threads = 8 waves.
//   Wave w handles K-steps s = w, w+8, ... (3500 steps of K=4 total;
//   14000 = 3500*4 and 875*16 exactly -> no tails, bounds checked below).
//
// A-fragment (ISA 7.12.2, 32-bit A 16x4): lane l holds row M=l&15,
//   VGPR p holds K = p + 2*(l>=16)  -> contiguous float2 load per lane.
// B-fragment (4x16, row-striped): VGPR p, lane l: K = p + 2*(l>=16), N=l&15.
//   Column N==0 carries V[k], column N==1 carries 1.0, rest 0 -- built
//   branchlessly via per-lane FMA selectors so the V load stays
//   unconditional (and scalar: uniform address via readfirstlane).
// C/D (16x16 f32): lane = (M>=8 ? 16:0)+N, VGPR = M&7
//   -> lanes 0,1,16,17 hold the two useful columns.
// ---------------------------------------------------------------------------
__device__ __forceinline__ void wmma_step(
    const float* __restrict__ WrowK,   // per-lane: row base + 2*half
    const float* __restrict__ V,       // uniform
    int s, int half, float sel_v, float sel_1, v8f& c) {
  const int kbase = __builtin_amdgcn_readfirstlane(s << 2);  // uniform

  // A: non-temporal float2 (W is streamed exactly once; don't pollute L2)
  v2f a = __builtin_nontemporal_load((const v2f*)(WrowK + kbase));

  // B: scalar broadcast of V[kbase..kbase+3], branchless column build.
  const float4 v4 = *(const float4*)(V + kbase);
  const float vk0 = half ? v4.z : v4.x;   // K = 0 + 2*half
  const float vk1 = half ? v4.w : v4.y;   // K = 1 + 2*half
  v2f b;
  b.x = __builtin_fmaf(sel_v, vk0, sel_1);
  b.y = __builtin_fmaf(sel_v, vk1, sel_1);

  c = __builtin_amdgcn_wmma_f32_16x16x4_f32(
      /*neg_a=*/false, a, /*neg_b=*/false, b,
      /*c_mod=*/(short)0, c, /*reuse_a=*/false, /*reuse_b=*/false);
}

__global__ __launch_bounds__(256) void gap_gemv_kernel(
    const float* __restrict__ W, const float* __restrict__ V,
    float* __restrict__ out_dot, float* __restrict__ out_gsum) {
  const int lane = threadIdx.x & 31;
  const int wave = threadIdx.x >> 5;
  const int m    = lane & 15;       // row within 16-row tile
  const int half = lane >> 4;       // 0: K={0,1}, 1: K={2,3}
  const int row_base = blockIdx.x * 16;

  const float* __restrict__ WrowK =
      W + (long long)(row_base + m) * (long long)N_PV + (half << 1);

  const float sel_v = (m == 0) ? 1.0f : 0.0f;   // column 0 <- V[k]
  const float sel_1 = (m == 1) ? 1.0f : 0.0f;   // column 1 <- 1.0

  v8f c0 = {0.f, 0.f, 0.f, 0.f, 0.f, 0.f, 0.f, 0.f};
  v8f c1 = {0.f, 0.f, 0.f, 0.f, 0.f, 0.f, 0.f, 0.f};

  const int nsteps = N_PV / 4;  // 3500
  int s = wave;
  // Two independent accumulator chains -> 2x WMMA ILP and paired loads;
  // unroll doubles that to ~4 A-loads in flight per wave.
  #pragma unroll 2
  for (; s + 8 < nsteps; s += 16) {
    wmma_step(WrowK, V, s,     half, sel_v, sel_1, c0);
    wmma_step(WrowK, V, s + 8, half, sel_v, sel_1, c1);
  }
  if (s < nsteps) wmma_step(WrowK, V, s, half, sel_v, sel_1, c0);

  const v8f c = c0 + c1;

  // Cross-wave reduction of the two useful columns.
  __shared__ float red[8][16][2];
  if ((lane & 14) == 0) {           // lanes 0,1,16,17
    const int which = lane & 1;     // 0 = dot (N=0), 1 = rowsum (N=1)
    const int mofs  = (lane >> 4) * 8;
    red[wave][mofs + 0][which] = c[0];
    red[wave][mofs + 1][which] = c[1];
    red[wave][mofs + 2][which] = c[2];
    red[wave][mofs + 3][which] = c[3];
    red[wave][mofs + 4][which] = c[4];
    red[wave][mofs + 5][which] = c[5];
    red[wave][mofs + 6][which] = c[6];
    red[wave][mofs + 7][which] = c[7];
  }
  __syncthreads();

  if (threadIdx.x < 32) {
    const int r     = threadIdx.x >> 1;
    const int which = threadIdx.x & 1;
    float acc = 0.0f;
    #pragma unroll
    for (int w = 0; w < 8; ++w) acc += red[w][r][which];
    if (which == 0) out_dot[row_base + r]  = acc;
    else            out_gsum[row_base + r] = acc;
  }
}

// ---------------------------------------------------------------------------
// Kernel 3: elementwise LIF step for all three populations.
// ---------------------------------------------------------------------------
struct LifParams {
  const float* g_exc; const float* g_inh;
  const float* V; const float* tau; const float* vth;
  const float* gL; const float* vreset; const float* w;
};

__device__ __forceinline__ float lif_step(
    float V, float g_ampa, float g_nmda, float g_gaba,
    float tau, float vth, float gL, float w, float g_gap, float E_gap) {
  float syn = (g_ampa * (E_E_C - V) + g_nmda * (E_NMDA_C - V)
             + g_gaba * (E_I_C - V) + g_gap * (E_gap - V) - w) / gL;
  float V_new = V + (1.0f / tau) * ((E_L_C - V) + syn);
  return (V_new >= vth) ? 1.0f : 0.0f;
}

__global__ __launch_bounds__(256) void lif_all_kernel(
    LifParams pv, LifParams olm, LifParams bist,
    const float* __restrict__ drive_p,
    const float* __restrict__ gap_dot,
    const float* __restrict__ gap_sum,
    float* __restrict__ out) {
  const int i = blockIdx.x * 256 + threadIdx.x;
  if (i >= N_OUT) return;

  if (i < N_PV) {
    const float drive  = drive_p[0];
    const float gain   = fminf(1.0f, drive * 10.0f);
    const float active = (drive > 0.01f) ? 1.0f : 0.0f;
    const float gsum   = gap_sum[i];
    const float E_gap  = gap_dot[i] / (gsum + 1e-9f);
    const float g_gap  = gsum * gain * active;
    const float ge = pv.g_exc[i];
    out[i] = lif_step(pv.V[i], ge * 0.7f, ge * 0.3f, pv.g_inh[i],
                      pv.tau[i], pv.vth[i], pv.gL[i], pv.w[i], g_gap, E_gap);
  } else if (i < N_PV + N_OLM) {
    const int j = i - N_PV;
    const float ge = olm.g_exc[j];
    out[i] = lif_step(olm.V[j], ge * 0.95f, ge * 0.05f, olm.g_inh[j],
                      olm.tau[j], olm.vth[j], olm.gL[j], olm.w[j], 0.0f, 0.0f);
  } else {
    const int j = i - N_PV - N_OLM;
    const float ge = bist.g_exc[j];
    out[i] = lif_step(bist.V[j], ge * 0.95f, ge * 0.05f, bist.g_inh[j],
                      bist.tau[j], bist.vth[j], bist.gL[j], bist.w[j], 0.0f, 0.0f);
  }
}

// ---------------------------------------------------------------------------
// Launch
// ---------------------------------------------------------------------------
extern "C" void kernel_launch(void* const* d_in, const int* in_sizes, int n_in,
                              void* d_out, int out_size, void* d_ws, size_t ws_size,
                              hipStream_t stream) {
  const float* pv_g_exc   = (const float*)d_in[0];
  const float* pv_g_inh   = (const float*)d_in[1];
  const float* olm_g_exc  = (const float*)d_in[2];
  const float* olm_g_inh  = (const float*)d_in[3];
  const float* bist_g_exc = (const float*)d_in[4];
  const float* bist_g_inh = (const float*)d_in[5];
  const float* W_gap      = (const float*)d_in[6];
  const float* pv_V       = (const float*)d_in[7];
  const float* pv_tau     = (const float*)d_in[8];
  const float* pv_vth     = (const float*)d_in[9];
  const float* pv_gL      = (const float*)d_in[10];
  const float* pv_vreset  = (const float*)d_in[11];
  const float* pv_w       = (const float*)d_in[12];
  const float* olm_V      = (const float*)d_in[13];
  const float* olm_tau    = (const float*)d_in[14];
  const float* olm_vth    = (const float*)d_in[15];
  const float* olm_gL     = (const float*)d_in[16];
  const float* olm_vreset = (const float*)d_in[17];
  const float* olm_w      = (const float*)d_in[18];
  const float* bist_V     = (const float*)d_in[19];
  const float* bist_tau   = (const float*)d_in[20];
  const float* bist_vth   = (const float*)d_in[21];
  const float* bist_gL    = (const float*)d_in[22];
  const float* bist_vreset= (const float*)d_in[23];
  const float* bist_w     = (const float*)d_in[24];

  // Workspace layout (floats): [0] drive, [16..16+14000) dot, then gsum.
  float* ws       = (float*)d_ws;
  float* ws_drive = ws;
  float* ws_dot   = ws + 16;
  float* ws_gsum  = ws + 16 + N_PV;

  reduce_drive_kernel<<<1, 256, 0, stream>>>(pv_g_exc, ws_drive);

  gap_gemv_kernel<<<N_PV / 16, 256, 0, stream>>>(W_gap, pv_V, ws_dot, ws_gsum);

  LifParams pv  { pv_g_exc,  pv_g_inh,  pv_V,  pv_tau,  pv_vth,  pv_gL,  pv_vreset,  pv_w  };
  LifParams olm { olm_g_exc, olm_g_inh, olm_V, olm_tau, olm_vth, olm_gL, olm_vreset, olm_w };
  LifParams bi  { bist_g_exc,bist_g_inh,bist_V,bist_tau,bist_vth,bist_gL,bist_vreset,bist_w};

  lif_all_kernel<<<(N_OUT + 255) / 256, 256, 0, stream>>>(
      pv, olm, bi, ws_drive, ws_dot, ws_gsum, (float*)d_out);

  (void)in_sizes; (void)n_in; (void)out_size; (void)ws_size;
}